// BanditLayer_19198503813586
// MI455X (gfx1250) — compile-verified
//
#include <hip/hip_runtime.h>

typedef __attribute__((ext_vector_type(2))) float v2f;
typedef __attribute__((ext_vector_type(4))) float v4f;
typedef __attribute__((ext_vector_type(8))) float v8f;

#define TILE_ROWS 128
#define TILE_COLS 64

// ---------------------------------------------------------------------------
// Kernel 1: out[n,m] = dot(x[n,:], w[m,:]) + bias[m] via V_WMMA_F32_16X16X4_F32.
// Block = 256 threads = 8 waves; block tile 128x64; wave tile 32x32 (4 accs).
// Full-K LDS staging: x panel 128xD + w panel 64xD live in LDS (D=256 ->
// ~195 KB of the WGP's 320 KB). Row stride padded to D+4 dwords so the
// fragment reads (lane l -> row l%16, dword 2*(l>>4)) hit 64 distinct banks.
// A/B frags then come from ds_load_b64 with trivially hideable latency
// instead of per-iteration global round trips.
// ---------------------------------------------------------------------------
__global__ __launch_bounds__(256)
void bandit_gemm_bias_kernel(const float* __restrict__ x,
                             const float* __restrict__ w,
                             const float* __restrict__ bias,
                             float* __restrict__ out,
                             int N, int M, int D) {
  extern __shared__ float lds[];
  const int XS = D + 4;                       // padded row stride (dwords)
  float* ldsX = lds;                          // TILE_ROWS x XS
  float* ldsW = lds + TILE_ROWS * XS;         // TILE_COLS x XS

  const int tid  = threadIdx.x;
  const int lane = tid & 31;
  const int wave = tid >> 5;
  const int l16  = lane & 15;
  const int lh   = lane >> 4;                 // 0 or 1 (K half of fragment)
  const int wy   = wave & 3;                  // 4 row groups of 32
  const int wx   = wave >> 2;                 // 2 col groups of 32
  const int rowBase = blockIdx.y * TILE_ROWS;
  const int colBase = blockIdx.x * TILE_COLS;

  // ---- stage x panel (TILE_ROWS x D) and w panel (TILE_COLS x D) into LDS
  const int c4 = D >> 2;                      // float4s per row
  for (int i = tid; i < TILE_ROWS * c4; i += 256) {
    const int r = i / c4, c = (i - r * c4) << 2;
    const v4f v = *(const v4f*)(x + (size_t)(rowBase + r) * D + c);
    *(v4f*)(ldsX + r * XS + c) = v;
  }
  for (int i = tid; i < TILE_COLS * c4; i += 256) {
    const int r = i / c4, c = (i - r * c4) << 2;
    const v4f v = *(const v4f*)(w + (size_t)(colBase + r) * D + c);
    *(v4f*)(ldsW + r * XS + c) = v;
  }
  __syncthreads();

  v8f acc00 = {}, acc01 = {}, acc10 = {}, acc11 = {};

  const float* pA0 = ldsX + (wy * 32 + l16) * XS + 2 * lh;
  const float* pA1 = pA0 + 16 * XS;
  const float* pB0 = ldsW + (wx * 32 + l16) * XS + 2 * lh;
  const float* pB1 = pB0 + 16 * XS;

#pragma unroll 8
  for (int k = 0; k < D; k += 4) {
    v2f a0 = *(const v2f*)(pA0 + k);
    v2f a1 = *(const v2f*)(pA1 + k);
    v2f b0 = *(const v2f*)(pB0 + k);
    v2f b1 = *(const v2f*)(pB1 + k);
    acc00 = __builtin_amdgcn_wmma_f32_16x16x4_f32(false, a0, false, b0,
                                                  (short)0, acc00, false, false);
    acc01 = __builtin_amdgcn_wmma_f32_16x16x4_f32(false, a0, false, b1,
                                                  (short)0, acc01, false, false);
    acc10 = __builtin_amdgcn_wmma_f32_16x16x4_f32(false, a1, false, b0,
                                                  (short)0, acc10, false, false);
    acc11 = __builtin_amdgcn_wmma_f32_16x16x4_f32(false, a1, false, b1,
                                                  (short)0, acc11, false, false);
  }

  // ---- epilogue: C/D layout vgpr r, lane l -> M = r + (l>>4)*8, N = l%16
  const int row0 = rowBase + wy * 32;
  const int col0 = colBase + wx * 32;
  const float bia0 = bias[col0 + l16];
  const float bia1 = bias[col0 + 16 + l16];
#pragma unroll
  for (int r = 0; r < 8; ++r) {
    const int mr = r + lh * 8;
    out[(size_t)(row0 + mr) * M + col0 + l16]           = acc00[r] + bia0;
    out[(size_t)(row0 + mr) * M + col0 + 16 + l16]      = acc01[r] + bia1;
    out[(size_t)(row0 + 16 + mr) * M + col0 + l16]      = acc10[r] + bia0;
    out[(size_t)(row0 + 16 + mr) * M + col0 + 16 + l16] = acc11[r] + bia1;
  }
}

// ---------------------------------------------------------------------------
// Kernel 2: per-column exact top-K mask. One 256-thread workgroup per column m.
// Column cached in LDS (N=16384 keys = 64 KB of the 320 KB WGP LDS).
// 4-pass MSB-first radix select on monotonic uint keys finds the K-th largest;
// ties broken in increasing row order (matches jax top_k). Non-kept -> 0.
// ---------------------------------------------------------------------------
__global__ __launch_bounds__(256)
void bandit_topk_mask_kernel(float* __restrict__ out,
                             const int* __restrict__ active_k,
                             int N, int M) {
  extern __shared__ unsigned smem[];
  unsigned* keys  = smem;           // [N]
  unsigned* hist  = smem + N;       // [256]
  unsigned* cnts  = hist + 256;     // [256]
  unsigned* state = cnts + 256;     // [0]=prefix, [1]=krem

  const int m   = blockIdx.x;
  const int tid = threadIdx.x;
  const int T   = blockDim.x;
  const unsigned K = (unsigned)active_k[0];

  // Load column, transform to order-preserving unsigned keys.
  for (int n = tid; n < N; n += T) {
    unsigned b = __float_as_uint(out[(size_t)n * M + m]);
    keys[n] = (b & 0x80000000u) ? ~b : (b | 0x80000000u);
  }
  if (tid == 0) { state[0] = 0u; state[1] = K; }
  __syncthreads();

  // Radix select (8 bits per pass, MSB first) for the K-th largest key.
  for (int pass = 0; pass < 4; ++pass) {
    const int shift = 24 - 8 * pass;
    const unsigned hmask = (pass == 0) ? 0u : (0xFFFFFFFFu << (shift + 8));
    if (tid < 256) hist[tid] = 0u;
    __syncthreads();
    const unsigned prefix = state[0];
    for (int n = tid; n < N; n += T) {
      const unsigned key = keys[n];
      if (((key ^ prefix) & hmask) == 0u)
        atomicAdd(&hist[(key >> shift) & 255u], 1u);
    }
    __syncthreads();
    if (tid == 0) {
      unsigned krem = state[1];
      unsigned cum = 0;
      for (int b = 255; b >= 0; --b) {
        const unsigned h = hist[b];
        if (cum + h >= krem) {
          state[0] = prefix | ((unsigned)b << shift);
          state[1] = krem - cum;
          break;
        }
        cum += h;
      }
    }
    __syncthreads();
  }

  const unsigned Tkey = state[0];   // exact K-th largest key
  const unsigned need = state[1];   // # of Tkey-valued ties to keep (index order)

  // Index-ordered tie ranking over contiguous per-thread chunks.
  const int chunk = (N + T - 1) / T;
  const int nlo = tid * chunk;
  int nhi = nlo + chunk; if (nhi > N) nhi = N;
  unsigned mycnt = 0;
  for (int n = nlo; n < nhi; ++n) mycnt += (keys[n] == Tkey) ? 1u : 0u;
  cnts[tid] = mycnt;
  __syncthreads();
  if (tid == 0) {
    unsigned run = 0;
    for (int t = 0; t < T; ++t) { const unsigned c = cnts[t]; cnts[t] = run; run += c; }
  }
  __syncthreads();

  unsigned rank = cnts[tid];
  for (int n = nlo; n < nhi; ++n) {
    const unsigned key = keys[n];
    bool keep;
    if (key > Tkey)        keep = true;
    else if (key == Tkey)  { keep = (rank < need); ++rank; }
    else                   keep = false;
    if (!keep) out[(size_t)n * M + m] = 0.0f;
  }
}

extern "C" void kernel_launch(void* const* d_in, const int* in_sizes, int n_in,
                              void* d_out, int out_size, void* d_ws, size_t ws_size,
                              hipStream_t stream) {
  const float* x        = (const float*)d_in[0];
  const float* w        = (const float*)d_in[1];
  const float* bias     = (const float*)d_in[2];
  const int*   active_k = (const int*)d_in[3];
  float* out = (float*)d_out;

  const int M = in_sizes[2];            // 4096
  const int D = in_sizes[1] / M;        // 256
  const int N = in_sizes[0] / D;        // 16384

  dim3 g1(M / TILE_COLS, N / TILE_ROWS);
  const size_t shmem1 = (size_t)(TILE_ROWS + TILE_COLS) * (D + 4) * sizeof(float);
  bandit_gemm_bias_kernel<<<g1, dim3(256), shmem1, stream>>>(x, w, bias, out, N, M, D);

  const size_t shmem2 = (size_t)(N + 256 + 256 + 8) * sizeof(unsigned);
  bandit_topk_mask_kernel<<<dim3(M), dim3(256), shmem2, stream>>>(out, active_k, N, M);
}